// ChamferLoss_75290776699474
// MI455X (gfx1250) — compile-verified
//
#include <hip/hip_runtime.h>
#include <hip/hip_bf16.h>
#include <math.h>

typedef __attribute__((ext_vector_type(2))) float v2f;
typedef __attribute__((ext_vector_type(8))) float v8f;

#define BATCH 16
#define NPTS  2048          // M == N == 2048
#define NT    (NPTS / 16)   // 128 tiles of 16
#define WAVES 8             // waves per block
#define MG    (NT / WAVES)  // 16 m-tile-groups per batch

// ---------------------------------------------------------------------------
// Kernel 1: init backward-min array to +inf bit pattern (u32 atomic-min ready)
// ---------------------------------------------------------------------------
__global__ __launch_bounds__(256) void chamfer_init_kernel(unsigned int* bwd) {
    int i = blockIdx.x * 256 + threadIdx.x;
    if (i < BATCH * NPTS) bwd[i] = 0x7F800000u;  // +inf
}

// ---------------------------------------------------------------------------
// Kernel 2: WMMA pairwise squared distances with norms folded into the GEMM.
//   A-row(m) = (-2*tx, -2*ty, |t|^2, 1)   (K = 0..3)
//   B-col(n) = ( ax,    ay,   1,  |a|^2)
//   => C[m][n] = |t_m - a_n|^2 directly.
// B operands are pre-expanded to float4 {ax, ay, 1, |a|^2} in LDS, so the
// inner loop is: ds_load_b128 (prefetched one tile ahead) -> dual cndmask ->
// wmma -> min3 chain -> ds_min_u32.
// ---------------------------------------------------------------------------
__global__ __launch_bounds__(256) void chamfer_wmma_kernel(
    const float2* __restrict__ target,   // [B, 2048] float2
    const float2* __restrict__ actual,   // [B, 2048] float2
    float* __restrict__ fwd,             // [B, 2048] per-target min d^2
    unsigned int* __restrict__ bwd)      // [B, 2048] per-actual min d^2 (bits)
{
    __shared__ float4       aLds[NPTS];        // 32 KB: {ax, ay, 1, |a|^2}
    __shared__ unsigned int bwdLds[NPTS];      //  8 KB: per-actual min (bits)

    const int tid  = threadIdx.x;
    const int b    = blockIdx.x >> 4;          // batch
    const int mg   = blockIdx.x & 15;          // m-tile group
    const int w    = tid >> 5;                 // wave in block
    const int lane = tid & 31;
    const int m_base = (mg * WAVES + w) * 16;
    const int nb = b * NPTS;

    // ---- stage expanded B operands into LDS; init LDS backward mins
    const float2* ab = actual + nb;
#pragma unroll
    for (int k = 0; k < NPTS / 256; ++k) {
        int i = tid + k * 256;
        float2 a = ab[i];
        aLds[i]   = make_float4(a.x, a.y, 1.0f, fmaf(a.y, a.y, a.x * a.x));
        bwdLds[i] = 0x7F800000u;
    }
    __syncthreads();

    // ---- A operand (once per wave): norms folded in
    const bool lowHalf = (lane < 16);
    float2 tpt = target[nb + m_base + (lane & 15)];
    float  tn  = fmaf(tpt.y, tpt.y, tpt.x * tpt.x);
    v2f avec;
    avec.x = lowHalf ? (-2.0f * tpt.x) : tn;     // K=0 / K=2
    avec.y = lowHalf ? (-2.0f * tpt.y) : 1.0f;   // K=1 / K=3

    float fmin2[8];
#pragma unroll
    for (int i = 0; i < 8; ++i) fmin2[i] = INFINITY;

    const int col = lane & 15;
    float4 cur = aLds[col];                      // prefetch tile 0

    for (int nt = 0; nt < NT; ++nt) {
        // prefetch next tile's B column (wraps; always in-bounds)
        float4 nxt = aLds[(((nt + 1) & (NT - 1)) << 4) + col];

        v2f bvec;
        bvec.x = lowHalf ? cur.x : cur.z;        // K=0 / K=2
        bvec.y = lowHalf ? cur.y : cur.w;        // K=1 / K=3

        v8f c = {};
        c = __builtin_amdgcn_wmma_f32_16x16x4_f32(
                false, avec, false, bvec, (short)0, c, false, false);

        float bmin = INFINITY;
#pragma unroll
        for (int i = 0; i < 8; ++i) {
            fmin2[i] = fminf(fmin2[i], c[i]);    // forward running min
            bmin     = fminf(bmin, c[i]);        // backward tile min
        }
        bmin = fmaxf(bmin, 0.0f);  // clamp so u32 ordering == float ordering
        // both halves hit the same slot -> min over all 16 rows of the tile
        atomicMin(&bwdLds[(nt << 4) + col], __float_as_uint(bmin));

        cur = nxt;
    }

    // ---- flush LDS backward mins to global (order-independent => determin.)
    __syncthreads();
#pragma unroll
    for (int k = 0; k < NPTS / 256; ++k) {
        int i = tid + k * 256;
        atomicMin(&bwd[nb + i], bwdLds[i]);
    }

    // ---- forward min: butterfly within each 16-lane half
#pragma unroll
    for (int i = 0; i < 8; ++i) {
#pragma unroll
        for (int s = 1; s < 16; s <<= 1)
            fmin2[i] = fminf(fmin2[i], __shfl_xor(fmin2[i], s, 32));
    }
    if ((lane & 15) == 0) {
        const int half8 = lowHalf ? 0 : 8;
#pragma unroll
        for (int i = 0; i < 8; ++i)
            fwd[nb + m_base + half8 + i] = fmaxf(fmin2[i], 0.0f);
    }
}

// ---------------------------------------------------------------------------
// Kernel 3: out = mean(sqrt(fwd)) + mean(sqrt(bwd)); both means over 32768,
// so out = (sum over all 65536 sqrt values) / 32768.
// ---------------------------------------------------------------------------
__global__ __launch_bounds__(256) void chamfer_reduce_kernel(
    const float* __restrict__ vals,   // fwd[32768] then bwd[32768] (same bits)
    float* __restrict__ out)
{
    __shared__ float sm[256];
    const int tid = threadIdx.x;
    float s = 0.0f;
    for (int i = tid; i < 2 * BATCH * NPTS; i += 256)
        s += sqrtf(fmaxf(vals[i], 0.0f));
    sm[tid] = s;
    __syncthreads();
#pragma unroll
    for (int step = 128; step > 0; step >>= 1) {
        if (tid < step) sm[tid] += sm[tid + step];
        __syncthreads();
    }
    if (tid == 0) out[0] = sm[0] * (1.0f / (float)(BATCH * NPTS));
}

// ---------------------------------------------------------------------------
extern "C" void kernel_launch(void* const* d_in, const int* in_sizes, int n_in,
                              void* d_out, int out_size, void* d_ws, size_t ws_size,
                              hipStream_t stream) {
    (void)in_sizes; (void)n_in; (void)out_size; (void)ws_size;
    const float2* target = (const float2*)d_in[0];
    const float2* actual = (const float2*)d_in[1];

    float*        fwd = (float*)d_ws;                              // 32768 f32
    unsigned int* bwd = (unsigned int*)((char*)d_ws + BATCH * NPTS * sizeof(float));

    // 1) init backward mins
    chamfer_init_kernel<<<(BATCH * NPTS + 255) / 256, 256, 0, stream>>>(bwd);

    // 2) WMMA pairwise sweep: 16 batches x 16 m-tile groups = 256 blocks
    chamfer_wmma_kernel<<<BATCH * MG, 256, 0, stream>>>(target, actual, fwd, bwd);

    // 3) final reduction to scalar
    chamfer_reduce_kernel<<<1, 256, 0, stream>>>(fwd, (float*)d_out);
}